// GIN_49409303773907
// MI455X (gfx1250) — compile-verified
//
#include <hip/hip_runtime.h>
#include <hip/hip_bf16.h>

#define N_NODES   50000
#define N_EDGES   800000
#define HID       128
#define OUT_DIM   40
#define N_LAYERS  4
#define N_GRAPHS  128
#define BN_EPS    1e-5f

#define WT_STRIDE 132   // padded LDS stride: 132 mod 64 = 4 -> conflict-free b64 reads

typedef __attribute__((ext_vector_type(2))) float v2f;
typedef __attribute__((ext_vector_type(8))) float v8f;

// ---------------------------------------------------------------- utilities
__global__ void fill_zero(float* __restrict__ p, int n) {
  int i = blockIdx.x * blockDim.x + threadIdx.x;
  if (i < n) p[i] = 0.0f;
}

// -------------------------------------------------- edge gather/scatter-add
// one edge handled by 32 threads; each thread adds a float4 column slice
__global__ void edge_agg(const float* __restrict__ x,
                         const int* __restrict__ ei,
                         float* __restrict__ agg) {
  long i = (long)blockIdx.x * blockDim.x + threadIdx.x;
  long e = i >> 5;
  if (e >= N_EDGES) return;
  int c = (int)(i & 31) * 4;
  int src = ei[e];
  int dst = ei[N_EDGES + e];
  const float4 v = *(const float4*)(x + (long)src * HID + c);
  float* d = agg + (long)dst * HID + c;
  atomicAdd(d + 0, v.x);
  atomicAdd(d + 1, v.y);
  atomicAdd(d + 2, v.z);
  atomicAdd(d + 3, v.w);
}

// ------------------------------------------------------- fused WMMA GEMM
// out[M,128] = epilogue( A[M,128] @ W[128,128] + bias )
// COMBINE: A element = (1+eps)*A0 + A1  (GIN combine fused into A-load)
// BN:      epilogue = relu then inference BatchNorm; else relu only
template <bool COMBINE, bool BN>
__global__ __launch_bounds__(256)
void gemm_fused(const float* __restrict__ A0,
                const float* __restrict__ A1,
                const float* __restrict__ eps_arr, int layer,
                const float* __restrict__ W,
                const float* __restrict__ bias,
                const float* __restrict__ gamma,
                const float* __restrict__ beta,
                const float* __restrict__ mean,
                const float* __restrict__ var,
                float* __restrict__ out, int nrows) {
  // W transposed in LDS with padded stride: B-fragment (W[k][n], W[k+1][n])
  // is contiguous -> single ds_load_b64, no register shuffling.
  __shared__ __align__(16) float Wt[HID * WT_STRIDE];  // 67,584 B
  for (int idx = threadIdx.x; idx < HID * HID; idx += blockDim.x) {
    int k = idx >> 7, n = idx & 127;
    Wt[n * WT_STRIDE + k] = W[idx];
  }
  __syncthreads();

  const int wave = threadIdx.x >> 5;
  const int lane = threadIdx.x & 31;
  const int strip = blockIdx.x * 8 + wave;
  if (strip * 16 >= nrows) return;      // uniform per-wave: EXEC stays all-1s

  const int m = lane & 15;              // row within strip (A), col within tile (B/C)
  const int half = lane >> 4;           // K sub-pair selector per ISA layout
  const long row = (long)strip * 16 + m;

  const float scale = COMBINE ? (1.0f + eps_arr[layer]) : 1.0f;
  const float* arow0 = A0 + row * HID;
  const float* arow1 = COMBINE ? (A1 + row * HID) : nullptr;
  // per-lane LDS base for B fragments: tile offsets are compile-time immediates
  const float* wtl = Wt + m * WT_STRIDE + half * 2;

  v8f acc[8] = {};                      // 8 tiles of 16x16 -> full 16x128 strip

  for (int kc = 0; kc < 32; ++kc) {     // K = 128 in chunks of 4
    const int k0 = kc * 4 + half * 2;
    // A fragment: lane holds A[m][k0], A[m][k0+1]
    v2f a = *(const v2f*)(arow0 + k0);
    if (COMBINE) {
      v2f g = *(const v2f*)(arow1 + k0);
      a.x = scale * a.x + g.x;
      a.y = scale * a.y + g.y;
    }
    const float* wk = wtl + kc * 4;
#pragma unroll
    for (int t = 0; t < 8; ++t) {
      // B fragment: W[k0][n], W[k0+1][n] contiguous in transposed LDS
      v2f b = *(const v2f*)(wk + t * 16 * WT_STRIDE);
      acc[t] = __builtin_amdgcn_wmma_f32_16x16x4_f32(
          false, a, false, b, (short)0, acc[t], false, false);
    }
  }

  // epilogue: C/D layout -> VGPR r holds row (r + half*8), col m of each tile
#pragma unroll
  for (int r = 0; r < 8; ++r) {
    const long orow = (long)strip * 16 + r + half * 8;
#pragma unroll
    for (int t = 0; t < 8; ++t) {
      const int n = t * 16 + m;
      float v = acc[t][r] + bias[n];
      v = fmaxf(v, 0.0f);
      if (BN)
        v = gamma[n] * (v - mean[n]) * rsqrtf(var[n] + BN_EPS) + beta[n];
      out[orow * HID + n] = v;
    }
  }
}

// ------------------------------------------------------------ mean pooling
__global__ void pool_sum(const float* __restrict__ x,
                         const int* __restrict__ batch,
                         float* __restrict__ sums,
                         float* __restrict__ counts) {
  long i = (long)blockIdx.x * blockDim.x + threadIdx.x;
  long node = i >> 5;
  if (node >= N_NODES) return;
  int q = (int)(i & 31);
  int g = batch[node];
  const float4 v = *(const float4*)(x + node * HID + q * 4);
  float* d = sums + (long)g * HID + q * 4;
  atomicAdd(d + 0, v.x);
  atomicAdd(d + 1, v.y);
  atomicAdd(d + 2, v.z);
  atomicAdd(d + 3, v.w);
  if (q == 0) atomicAdd(counts + g, 1.0f);
}

// ------------------------------------------- classification head (tiny)
__global__ __launch_bounds__(128)
void head(const float* __restrict__ sums, const float* __restrict__ counts,
          const float* __restrict__ lin1W, const float* __restrict__ lin1b,
          const float* __restrict__ lin2W, const float* __restrict__ lin2b,
          float* __restrict__ out) {
  __shared__ float p[HID];
  __shared__ float h1[HID];
  __shared__ float lg[OUT_DIM];
  __shared__ float red[2];
  const int g = blockIdx.x, c = threadIdx.x;

  const float cnt = fmaxf(counts[g], 1.0f);
  p[c] = sums[(long)g * HID + c] / cnt;
  __syncthreads();

  float acc = lin1b[c];
  for (int k = 0; k < HID; ++k) acc += p[k] * lin1W[k * HID + c];
  h1[c] = fmaxf(acc, 0.0f);
  __syncthreads();

  if (c < OUT_DIM) {
    float a = lin2b[c];
    for (int k = 0; k < HID; ++k) a += h1[k] * lin2W[k * OUT_DIM + c];
    lg[c] = a;
  }
  __syncthreads();

  if (c == 0) {
    float mx = lg[0];
    for (int k = 1; k < OUT_DIM; ++k) mx = fmaxf(mx, lg[k]);
    float s = 0.0f;
    for (int k = 0; k < OUT_DIM; ++k) s += expf(lg[k] - mx);
    red[0] = mx;
    red[1] = logf(s);
  }
  __syncthreads();

  if (c < OUT_DIM) out[(long)g * OUT_DIM + c] = lg[c] - red[0] - red[1];
}

// ----------------------------------------------------------------- launcher
extern "C" void kernel_launch(void* const* d_in, const int* in_sizes, int n_in,
                              void* d_out, int out_size, void* d_ws, size_t ws_size,
                              hipStream_t stream) {
  const float* x0     = (const float*)d_in[0];
  const int*   ei     = (const int*)d_in[1];
  const int*   batch  = (const int*)d_in[2];
  const float* W1s    = (const float*)d_in[3];
  const float* b1s    = (const float*)d_in[4];
  const float* W2s    = (const float*)d_in[5];
  const float* b2s    = (const float*)d_in[6];
  const float* gammas = (const float*)d_in[7];
  const float* betas  = (const float*)d_in[8];
  const float* means  = (const float*)d_in[9];
  const float* vars   = (const float*)d_in[10];
  const float* epsarr = (const float*)d_in[11];
  const float* lin1W  = (const float*)d_in[12];
  const float* lin1b  = (const float*)d_in[13];
  const float* lin2W  = (const float*)d_in[14];
  const float* lin2b  = (const float*)d_in[15];
  float* out = (float*)d_out;

  const long NF = (long)N_NODES * HID;  // 6,400,000 floats
  float* ws   = (float*)d_ws;
  float* agg  = ws;            // node buffer 0
  float* t1   = ws + NF;       // node buffer 1 (free after last gemm2 -> reuse for pool)
  float* xb   = ws + 2 * NF;   // node buffer 2 (layer output / next input)

  const int ZB   = (int)((NF + 255) / 256);
  const int EB   = (int)(((long)N_EDGES * 32 + 255) / 256);
  const int PB   = (int)(((long)N_NODES * 32 + 255) / 256);
  const int GEMB = (N_NODES / 16 + 7) / 8;   // 3125 strips / 8 waves = 391 blocks

  const float* x = x0;
  for (int i = 0; i < N_LAYERS; ++i) {
    fill_zero<<<ZB, 256, 0, stream>>>(agg, (int)NF);
    edge_agg <<<EB, 256, 0, stream>>>(x, ei, agg);
    // GEMM1: relu(((1+eps)*x + agg) @ W1 + b1) -> t1
    gemm_fused<true, false><<<GEMB, 256, 0, stream>>>(
        x, agg, epsarr, i,
        W1s + (long)i * HID * HID, b1s + (long)i * HID,
        nullptr, nullptr, nullptr, nullptr,
        t1, N_NODES);
    // GEMM2: BN(relu(t1 @ W2 + b2)) -> xb
    gemm_fused<false, true><<<GEMB, 256, 0, stream>>>(
        t1, nullptr, nullptr, 0,
        W2s + (long)i * HID * HID, b2s + (long)i * HID,
        gammas + (long)i * HID, betas + (long)i * HID,
        means + (long)i * HID, vars + (long)i * HID,
        xb, N_NODES);
    x = xb;
  }

  // pooling scratch lives in the now-free t1 buffer
  float* sums   = t1;
  float* counts = t1 + (long)N_GRAPHS * HID;
  fill_zero<<<(N_GRAPHS * HID + N_GRAPHS + 255) / 256, 256, 0, stream>>>(
      sums, N_GRAPHS * HID + N_GRAPHS);
  pool_sum<<<PB, 256, 0, stream>>>(x, batch, sums, counts);
  head<<<N_GRAPHS, 128, 0, stream>>>(sums, counts, lin1W, lin1b, lin2W, lin2b, out);
}